// LocalAttention1d_80401787781566
// MI455X (gfx1250) — compile-verified
//
#include <hip/hip_runtime.h>
#include <hip/hip_bf16.h>
#include <math.h>

// Problem constants (from reference setup_inputs)
#define Bn   64
#define Qd   1024
#define Nn   2048
#define Cn   2048
#define Pn   512
#define Wn   256
#define SIZE (Nn + 1)   // 2049

typedef __attribute__((ext_vector_type(2))) float v2f;
typedef __attribute__((ext_vector_type(8))) float v8f;

// ---------------------------------------------------------------------------
// Kernel 1: H[b,p] = sum_c c_t[b,c] * w_p[p,c]   (64x512 = c_t @ w_p^T)
// One wave per 16x16 output tile, V_WMMA_F32_16X16X4_F32, K-loop over 2048.
// A-layout (16x4 f32): lane l<16 -> A[M=l][K=0],A[M=l][K=1] in vgpr0/1;
//                      lane l>=16 -> K=2,3. B mirrored with N in lanes.
// ---------------------------------------------------------------------------
__global__ __launch_bounds__(32)
void k_proj_wmma(const float* __restrict__ c_t,   // (B, C)
                 const float* __restrict__ w_p,   // (P, C)
                 float* __restrict__ H)           // (B, P)
{
    const int tile = blockIdx.x;          // 0 .. (B/16)*(P/16)-1 = 127
    const int tm   = tile / (Pn / 16);    // 0..3
    const int tn   = tile % (Pn / 16);    // 0..31
    const int lane = threadIdx.x & 31;
    const int l16  = lane & 15;
    const int hi   = lane >> 4;           // 0 or 1

    const float* arow = c_t + (size_t)(tm * 16 + l16) * Cn;  // A row (M = l16)
    const float* brow = w_p + (size_t)(tn * 16 + l16) * Cn;  // B col (N = l16), w_p row

    v8f acc = {};
    for (int k = 0; k < Cn; k += 4) {
        v2f a = *(const v2f*)(arow + k + 2 * hi);   // K = k+2*hi, k+2*hi+1
        v2f b = *(const v2f*)(brow + k + 2 * hi);
        acc = __builtin_amdgcn_wmma_f32_16x16x4_f32(
            /*neg_a=*/false, a, /*neg_b=*/false, b,
            /*c_mod=*/(short)0, acc, /*reuse_a=*/false, /*reuse_b=*/false);
    }

    // C/D layout: vgpr r, lane<16 -> (M=r, N=lane); lane>=16 -> (M=r+8, N=lane-16)
    const int m_base = tm * 16 + hi * 8;
    const int n      = tn * 16 + l16;
#pragma unroll
    for (int r = 0; r < 8; ++r)
        H[(size_t)(m_base + r) * Pn + n] = acc[r];
}

// ---------------------------------------------------------------------------
// Kernel 2: per-batch scalars + window weights.
//   logit = sum_p tanh(H[b,p]) * v_p[p];  loc = sigmoid(logit)
//   p_t = loc*(SIZE-2); p = rint(p_t)  (RNE, matches jnp.round)
//   per w: s_i = mod(clip(p + (w-127), 0, SIZE), SIZE); n = s_i-1
//          g = (s_i>=1) ? exp(-2*((n - p_t)/128)^2) : 0   [softmax(singleton)=1]
// ---------------------------------------------------------------------------
__global__ __launch_bounds__(256)
void k_params(const float* __restrict__ H,     // (B, P)
              const float* __restrict__ v_p,   // (P)
              float* __restrict__ G,           // (B, W)
              int* __restrict__ NC)            // (B, W)
{
    const int b = blockIdx.x;
    const int t = threadIdx.x;                 // 0..255

    __shared__ float red[256];
    __shared__ float s_pt;
    __shared__ int   s_p;

    float part = tanhf(H[(size_t)b * Pn + t])       * v_p[t] +
                 tanhf(H[(size_t)b * Pn + t + 256]) * v_p[t + 256];
    red[t] = part;
    __syncthreads();
#pragma unroll
    for (int s = 128; s > 0; s >>= 1) {
        if (t < s) red[t] += red[t + s];
        __syncthreads();
    }
    if (t == 0) {
        float logit = red[0];
        float loc   = 1.0f / (1.0f + expf(-logit));
        float pt    = loc * (float)(SIZE - 2);
        s_pt = pt;
        s_p  = (int)rintf(pt);   // round-to-nearest-even
    }
    __syncthreads();

    const float pt = s_pt;
    const int   p  = s_p;

    const int off = t - (Wn / 2 - 1);          // -127 .. 128
    int sp = p + off;
    sp = sp < 0 ? 0 : (sp > SIZE ? SIZE : sp); // clip [0, SIZE]
    const int si = (sp == SIZE) ? 0 : sp;      // mod SIZE
    const int n  = si - 1;
    const int nn = n < 0 ? 0 : n;              // clip(s_i-1, 0)
    const float d = ((float)nn - pt) * (1.0f / (float)(Wn / 2));
    const float g = (si >= 1) ? expf(-2.0f * d * d) : 0.0f;

    G[b * Wn + t]  = g;
    NC[b * Wn + t] = (si >= 1) ? n : -1;
}

// ---------------------------------------------------------------------------
// Kernel 3: s_t[b,q] = sum_w G[b,w] * q_i[b, q, NC[b,w]]   (memory-bound, 64MB)
// One wave32 per output row: lanes stride the 256-wide contiguous window
// (coalesced 128B requests), weights from LDS, shfl_xor wave reduction.
// 8 waves per block share one batch's weights (rows in a block share b).
// ---------------------------------------------------------------------------
__global__ __launch_bounds__(256)
void k_attend(const float* __restrict__ q_i,   // (B, Q, N)
              const float* __restrict__ G,     // (B, W)
              const int* __restrict__ NC,      // (B, W)
              float* __restrict__ out)         // (B, Q)
{
    const int lane = threadIdx.x & 31;
    const int wv   = threadIdx.x >> 5;                 // 0..7
    const int row  = blockIdx.x * 8 + wv;              // 0 .. B*Q-1
    const int b    = row / Qd;                         // constant per block
    const int q    = row % Qd;

    __shared__ float sg[Wn];
    __shared__ int   sn[Wn];
    {
        const int t = threadIdx.x;                     // blockDim == Wn == 256
        sg[t] = G[b * Wn + t];
        sn[t] = NC[b * Wn + t];
    }
    __syncthreads();

    const float* rowp = q_i + ((size_t)b * Qd + q) * Nn;
    float acc = 0.0f;
#pragma unroll
    for (int i = 0; i < Wn; i += 32) {
        const int w = i + lane;
        const int n = sn[w];
        const float val = (n >= 0) ? rowp[n] : 0.0f;   // contiguous across lanes
        acc = fmaf(sg[w], val, acc);
    }
#pragma unroll
    for (int s = 16; s > 0; s >>= 1)
        acc += __shfl_xor(acc, s, 32);
    if (lane == 0) out[row] = acc;
}

// ---------------------------------------------------------------------------
extern "C" void kernel_launch(void* const* d_in, const int* in_sizes, int n_in,
                              void* d_out, int out_size, void* d_ws, size_t ws_size,
                              hipStream_t stream) {
    // setup_inputs order: q_i, c_t, w_a, w_p, v_p, window
    const float* q_i = (const float*)d_in[0];
    const float* c_t = (const float*)d_in[1];
    // d_in[2] = w_a: provably unused (softmax over singleton axis == 1)
    const float* w_p = (const float*)d_in[3];
    const float* v_p = (const float*)d_in[4];
    (void)in_sizes; (void)n_in; (void)out_size; (void)ws_size;

    // Workspace layout: H (64*512 f32) | G (64*256 f32) | NC (64*256 i32) = 256KB
    float* H  = (float*)d_ws;
    float* G  = H + Bn * Pn;
    int*   NC = (int*)(G + Bn * Wn);

    k_proj_wmma<<<(Bn / 16) * (Pn / 16), 32, 0, stream>>>(c_t, w_p, H);
    k_params  <<<Bn, 256, 0, stream>>>(H, v_p, G, NC);
    k_attend  <<<(Bn * Qd) / 8, 256, 0, stream>>>(q_i, G, NC, (float*)d_out);
}